// NewPool2_20839181320913
// MI455X (gfx1250) — compile-verified
//
#include <hip/hip_runtime.h>
#include <hip/hip_bf16.h>

// Hyperbolic GNN layer for MI455X (gfx1250, wave32).
// Dominant cost: 3x edge scatter-adds (L2 atomics, tables L2-resident in 192MB).
// Dense GEMM (x_tan @ W_up) runs on the matrix pipe via V_WMMA_F32_16X16X4_F32,
// keeping exact f32 semantics of the reference.

typedef float v2f __attribute__((ext_vector_type(2)));
typedef float v8f __attribute__((ext_vector_type(8)));

#define FIN  128
#define FOUT 64

__device__ __forceinline__ float wave32_sum(float v) {
    #pragma unroll
    for (int m = 16; m >= 1; m >>= 1) v += __shfl_xor(v, m, 32);
    return v;
}

// ---------------------------------------------------------------------------
// K1: per-row logmap0 scale: s[i] = artanh(min(||x_i||,1-1e-7)) / max(||x_i||,1e-15)
// One wave32 per node, 128 floats = 32 float4 (one per lane).
__global__ __launch_bounds__(256) void k_logmap_scale(const float* __restrict__ x,
                                                      float* __restrict__ scale, int n) {
    int wave = threadIdx.x >> 5, lane = threadIdx.x & 31;
    int node = blockIdx.x * 8 + wave;
    if (node >= n) return;
    const float4* xr = (const float4*)(x + (size_t)node * FIN);
    float4 v = xr[lane];
    float ss = wave32_sum(v.x * v.x + v.y * v.y + v.z * v.z + v.w * v.w);
    if (lane == 0) {
        float nn = sqrtf(ss);
        float nd = fmaxf(nn, 1e-15f);            // clip(norm, MIN_NORM)
        float ca = fminf(nd, 1.0f - 1e-7f);      // _artanh input clip
        float at = 0.5f * (log1pf(ca) - log1pf(-ca));
        scale[node] = at / nd;
    }
}

// ---------------------------------------------------------------------------
// K2: zero the three accumulator buffers (4 floats per thread).
__global__ void k_zero(float* __restrict__ p, size_t n4) {
    size_t i = ((size_t)blockIdx.x * blockDim.x + threadIdx.x) * 4;
    if (i + 3 < n4) { p[i] = 0.f; p[i+1] = 0.f; p[i+2] = 0.f; p[i+3] = 0.f; }
    else { for (size_t j = i; j < n4; ++j) p[j] = 0.f; }
}

// ---------------------------------------------------------------------------
// K3: updated_x = leaky_relu( (x * s[row]) @ W_up ), via V_WMMA_F32_16X16X4_F32.
// Block = 256 threads = 8 waves; each wave computes a 16x64 output tile.
// W_up staged in LDS with K-pair interleaving so each B fragment is one b64 load:
//   ldsW[(k>>1)*128 + col*2 + (k&1)] = W_up[k*64 + col]
// A-frag layout (16x4 f32): lanes 0-15 -> K={4k,4k+1}, lanes 16-31 -> K={4k+2,4k+3}.
__global__ __launch_bounds__(256) void k_gemm_wmma(const float* __restrict__ x,
                                                   const float* __restrict__ scale,
                                                   const float* __restrict__ Wup,
                                                   float* __restrict__ upd, int n) {
    __shared__ float ldsW[FIN * FOUT];           // 32 KB of 320 KB WGP LDS
    for (int i = threadIdx.x; i < FIN * FOUT; i += 256) {
        int k = i >> 6, col = i & 63;
        ldsW[(k >> 1) * 128 + col * 2 + (k & 1)] = Wup[i];
    }
    __syncthreads();

    int wave = threadIdx.x >> 5, lane = threadIdx.x & 31;
    int row0 = blockIdx.x * 128 + wave * 16;
    int r    = row0 + (lane & 15);
    int khalf = (lane >> 4) * 2;                 // 0 for lanes 0-15, 2 for 16-31
    bool valid = (r < n);
    float sr = valid ? scale[r] : 0.0f;          // sr==0 zeroes A for tail rows
    const float* xrow = x + (size_t)(valid ? r : 0) * FIN;
    int colb = lane & 15;

    v8f c0 = {}, c1 = {}, c2 = {}, c3 = {};
    #pragma unroll 4
    for (int ks = 0; ks < FIN / 4; ++ks) {
        int kbase = ks * 4 + khalf;
        v2f a = *(const v2f*)(xrow + kbase);     // 8B aligned (kbase even)
        a.x *= sr; a.y *= sr;
        const v2f* wv = (const v2f*)&ldsW[(kbase >> 1) * 128];
        v2f b0 = wv[ 0 + colb];
        v2f b1 = wv[16 + colb];
        v2f b2 = wv[32 + colb];
        v2f b3 = wv[48 + colb];
        c0 = __builtin_amdgcn_wmma_f32_16x16x4_f32(false, a, false, b0, (short)0, c0, false, false);
        c1 = __builtin_amdgcn_wmma_f32_16x16x4_f32(false, a, false, b1, (short)0, c1, false, false);
        c2 = __builtin_amdgcn_wmma_f32_16x16x4_f32(false, a, false, b2, (short)0, c2, false, false);
        c3 = __builtin_amdgcn_wmma_f32_16x16x4_f32(false, a, false, b3, (short)0, c3, false, false);
    }

    // D layout: VGPR i holds M=i (lanes 0-15) / M=8+i (lanes 16-31), N=lane&15.
    int rhi = (lane >> 4) * 8;
    #pragma unroll
    for (int i = 0; i < 8; ++i) {
        int rr = row0 + rhi + i;
        if (rr < n) {
            float* o = upd + (size_t)rr * FOUT + colb;
            float v0 = c0[i], v1 = c1[i], v2 = c2[i], v3 = c3[i];
            o[ 0] = v0 > 0.f ? v0 : 0.01f * v0;  // leaky_relu slope 0.01
            o[16] = v1 > 0.f ? v1 : 0.01f * v1;
            o[32] = v2 > 0.f ? v2 : 0.01f * v2;
            o[48] = v3 > 0.f ? v3 : 0.01f * v3;
        }
    }
}

// ---------------------------------------------------------------------------
// K4/K6/K8: edge scatter-add: out[dst[e]] += w(src[e]) * updated_x[src[e]].
// One thread per (edge, 4 features): float4 gather + 4 global f32 atomics.
// mode 0: w=1   mode 1: w=sel[s]   mode 2: w=sel[s]*wsel[s]  (skip w==0 edges)
__global__ void k_scatter(const float* __restrict__ upd, const int* __restrict__ src,
                          const int* __restrict__ dst, const float* __restrict__ sel,
                          const float* __restrict__ wsel, float* __restrict__ out,
                          int mode, int E) {
    long long idx = (long long)blockIdx.x * 256 + threadIdx.x;
    if (idx >= (long long)E * 16) return;
    int e = (int)(idx >> 4);
    int f = ((int)idx & 15) * 4;
    int s = src[e];
    float w = 1.0f;
    if (mode >= 1) {
        w = sel[s];
        if (w == 0.0f) return;                   // hard gate prunes most atomics
        if (mode == 2) w *= wsel[s];
    }
    int d = dst[e];
    float4 v = *(const float4*)(upd + (size_t)s * FOUT + f);
    float* o = out + (size_t)d * FOUT + f;
    atomicAdd(o + 0, w * v.x);
    atomicAdd(o + 1, w * v.y);
    atomicAdd(o + 2, w * v.z);
    atomicAdd(o + 3, w * v.w);
}

// ---------------------------------------------------------------------------
// K5: gate: p = softmax(relu(sum_neigh @ W_pl)); sel = p[1] > 0.48
__global__ void k_gate(const float* __restrict__ sn, const float* __restrict__ Wpl,
                       float* __restrict__ sel, int n) {
    int i = blockIdx.x * 256 + threadIdx.x;
    if (i >= n) return;
    const float* s = sn + (size_t)i * FOUT;
    float d0 = 0.f, d1 = 0.f;
    #pragma unroll 8
    for (int f = 0; f < FOUT; ++f) { float v = s[f]; d0 += v * Wpl[2*f]; d1 += v * Wpl[2*f+1]; }
    d0 = fmaxf(d0, 0.f); d1 = fmaxf(d1, 0.f);
    float m = fmaxf(d0, d1);
    float e0 = expf(d0 - m), e1 = expf(d1 - m);
    sel[i] = (e1 / (e0 + e1) > 0.48f) ? 1.0f : 0.0f;
}

// ---------------------------------------------------------------------------
// K7: weight_sel = sigmoid(concat(sum_sel, sum_neigh) @ W_lw)
__global__ void k_wsel(const float* __restrict__ ssel, const float* __restrict__ sn,
                       const float* __restrict__ Wlw, float* __restrict__ wsel, int n) {
    int i = blockIdx.x * 256 + threadIdx.x;
    if (i >= n) return;
    const float* a = ssel + (size_t)i * FOUT;
    const float* b = sn   + (size_t)i * FOUT;
    float z = 0.f;
    #pragma unroll 8
    for (int f = 0; f < FOUT; ++f) z += a[f] * Wlw[f] + b[f] * Wlw[FOUT + f];
    wsel[i] = 1.0f / (1.0f + expf(-z));
}

// ---------------------------------------------------------------------------
// K9: out = proj(expmap0(updated_x + relu(a_x))). One wave32 per node (64 floats
// = 32 float2). ||y|| = tanh(n); proj clamps norm at (1 - 4e-3).
__global__ __launch_bounds__(256) void k_expmap_out(const float* __restrict__ upd,
                                                    const float* __restrict__ ax,
                                                    float* __restrict__ out, int n) {
    int wave = threadIdx.x >> 5, lane = threadIdx.x & 31;
    int node = blockIdx.x * 8 + wave;
    if (node >= n) return;
    const float2* ur = (const float2*)(upd + (size_t)node * FOUT);
    const float2* ar = (const float2*)(ax  + (size_t)node * FOUT);
    float2 u = ur[lane], a = ar[lane];
    u.x += fmaxf(a.x, 0.f);
    u.y += fmaxf(a.y, 0.f);
    float ss = wave32_sum(u.x * u.x + u.y * u.y);
    float nn = sqrtf(ss);
    float nc = fmaxf(nn, 1e-15f);
    float t  = tanhf(nc);
    float fac = t / nc;                          // expmap0 factor
    float ny  = fac * nn;                        // ||expmap0(u)||
    const float maxn = 1.0f - 4e-3f;             // proj boundary (c=1)
    if (ny > maxn) fac = maxn / nn;
    float2* o = (float2*)(out + (size_t)node * FOUT);
    float2 r; r.x = u.x * fac; r.y = u.y * fac;
    o[lane] = r;
}

// ---------------------------------------------------------------------------
extern "C" void kernel_launch(void* const* d_in, const int* in_sizes, int n_in,
                              void* d_out, int out_size, void* d_ws, size_t ws_size,
                              hipStream_t stream) {
    const float* x   = (const float*)d_in[0];   // [N,128]
    const int*   ei  = (const int*)  d_in[1];   // [2,E]
    const float* Wup = (const float*)d_in[2];   // [128,64]
    const float* Wpl = (const float*)d_in[3];   // [64,2]
    const float* Wlw = (const float*)d_in[4];   // [128,1]
    int N = in_sizes[0] / FIN;
    int E = in_sizes[1] / 2;
    const int* src = ei;
    const int* dst = ei + E;

    // Workspace layout (floats): upd | sum_neigh | sum_sel | a_x | scale | sel | wsel
    float* ws   = (float*)d_ws;
    float* upd  = ws;
    float* sumn = upd  + (size_t)N * FOUT;
    float* sums = sumn + (size_t)N * FOUT;
    float* axb  = sums + (size_t)N * FOUT;
    float* scl  = axb  + (size_t)N * FOUT;
    float* sel  = scl + N;
    float* wsl  = sel + N;
    float* out  = (float*)d_out;

    k_logmap_scale<<<(N + 7) / 8, 256, 0, stream>>>(x, scl, N);

    size_t zn = (size_t)N * FOUT * 3;            // sum_neigh + sum_sel + a_x
    k_zero<<<(unsigned)((zn + 1023) / 1024), 256, 0, stream>>>(sumn, zn);

    k_gemm_wmma<<<(N + 127) / 128, 256, 0, stream>>>(x, scl, Wup, upd, N);

    unsigned scb = (unsigned)(((long long)E * 16 + 255) / 256);
    k_scatter<<<scb, 256, 0, stream>>>(upd, src, dst, nullptr, nullptr, sumn, 0, E);
    k_gate<<<(N + 255) / 256, 256, 0, stream>>>(sumn, Wpl, sel, N);
    k_scatter<<<scb, 256, 0, stream>>>(upd, src, dst, sel, nullptr, sums, 1, E);
    k_wsel<<<(N + 255) / 256, 256, 0, stream>>>(sums, sumn, Wlw, wsl, N);
    k_scatter<<<scb, 256, 0, stream>>>(upd, src, dst, sel, wsl, axb, 2, E);
    k_expmap_out<<<(N + 7) / 8, 256, 0, stream>>>(upd, axb, out, N);
}